// KVCache_46583215293063
// MI455X (gfx1250) — compile-verified
//
#include <hip/hip_runtime.h>
#include <hip/hip_bf16.h>

// ---------------------------------------------------------------------------
// KV-cache update + gather (GQA repeat) for MI455X / gfx1250.
// Pure bandwidth problem (~1.1 GB moved -> ~47us floor @ 23.3 TB/s).
// Bulk cache copy uses the CDNA5 async global<->LDS DMA path
// (GLOBAL_LOAD_ASYNC_TO_LDS_B128 / GLOBAL_STORE_ASYNC_FROM_LDS_B128,
// ASYNCcnt + s_wait_asynccnt), per CDNA5 ISA.
// Each thread moves 64B per tensor (4 x b128 with immediate offsets) so the
// index decode is amortized and instruction issue stays far below the
// HBM-bandwidth limit.
// ---------------------------------------------------------------------------

namespace {
constexpr int kBSZ    = 2;
constexpr int kMaxSeq = 4096;
constexpr int kNKV    = 8;
constexpr int kHD     = 128;
}  // namespace

#ifndef __has_builtin
#define __has_builtin(x) 0
#endif

// Builtin parameter type (from clang diagnostic): int4 with address space.
typedef int v4i __attribute__((__vector_size__(4 * sizeof(int))));
typedef v4i __attribute__((address_space(1)))* gv4i_p;   // global int4*
typedef v4i __attribute__((address_space(3)))* lv4i_p;   // LDS int4*

#if __has_builtin(__builtin_amdgcn_global_load_async_to_lds_b128) && \
    __has_builtin(__builtin_amdgcn_global_store_async_from_lds_b128)
#define KV_USE_ASYNC_BUILTINS 1
#else
#define KV_USE_ASYNC_BUILTINS 0
#endif

// INST_OFFSET applies to BOTH the global and LDS addresses for async LDS ops
// (CDNA5 ISA), so one base pointer per side is enough.
template <int OFF>
__device__ __forceinline__ void async_g2l_b128(const void* g, void* l) {
#if KV_USE_ASYNC_BUILTINS
  __builtin_amdgcn_global_load_async_to_lds_b128((gv4i_p)g, (lv4i_p)l, OFF, 0);
#else
  unsigned loff = (unsigned)(unsigned long long)(lv4i_p)l + OFF;
  asm volatile("global_load_async_to_lds_b128 %0, %1, off"
               :: "v"(loff), "v"((unsigned long long)g + OFF)
               : "memory");
#endif
}

template <int OFF>
__device__ __forceinline__ void async_l2g_b128(void* g, void* l) {
#if KV_USE_ASYNC_BUILTINS
  __builtin_amdgcn_global_store_async_from_lds_b128((gv4i_p)g, (lv4i_p)l, OFF, 0);
#else
  unsigned loff = (unsigned)(unsigned long long)(lv4i_p)l + OFF;
  asm volatile("global_store_async_from_lds_b128 %0, %1, off"
               :: "v"((unsigned long long)g + OFF), "v"(loff)
               : "memory");
#endif
}

__device__ __forceinline__ void wait_async0() {
#if __has_builtin(__builtin_amdgcn_s_wait_asynccnt)
  __builtin_amdgcn_s_wait_asynccnt(0);
#else
  asm volatile("s_wait_asynccnt 0" ::: "memory");
#endif
}

// ---------------------------------------------------------------------------
// Kernel A: full cache -> k_new/v_new copy with the update slice
// [layer_idx, :, curr_pos : curr_pos+seq_len] sourced from xk/xv.
// One thread per 64B quad-chunk per tensor; movement via async DMA via LDS.
// ---------------------------------------------------------------------------
__global__ void __launch_bounds__(256)
kv_cache_writeback(const unsigned short* __restrict__ kc,
                   const unsigned short* __restrict__ vc,
                   const unsigned short* __restrict__ xk,
                   const unsigned short* __restrict__ xv,
                   const int* __restrict__ Lp,
                   const int* __restrict__ Pp,
                   int seq_len,
                   unsigned short* __restrict__ out_knew,
                   unsigned short* __restrict__ out_vnew,
                   int nQuads) {
  // 256 threads * (64B K + 64B V) = 32 KB LDS per block.
  __shared__ __align__(16) unsigned char smem[256 * 128];
  const int q = blockIdx.x * 256 + threadIdx.x;
  if (q >= nQuads) return;

  const int L = *Lp;
  const int P = *Pp;

  // Decode (layer, batch, t, head, d-quad): head row = 256B = 4 quads.
  const int dq = q & 3;                       // 64B quad within head row
  const int h  = (q >> 2) & (kNKV - 1);
  const int t  = (q >> 5) & (kMaxSeq - 1);
  const int b  = (q >> 17) & (kBSZ - 1);
  const int l  = q >> 18;

  const unsigned short* sK;
  const unsigned short* sV;
  if (l == L && t >= P && t < P + seq_len) {
    const size_t xo =
        ((((size_t)b * seq_len + (size_t)(t - P)) * kNKV + h) * kHD) +
        (size_t)dq * 32;
    sK = xk + xo;
    sV = xv + xo;
  } else {
    sK = kc + (size_t)q * 32;
    sV = vc + (size_t)q * 32;
  }

  unsigned char* mk = &smem[(unsigned)threadIdx.x * 128];
  unsigned char* mv = mk + 64;

  async_g2l_b128<0>(sK, mk);
  async_g2l_b128<16>(sK, mk);
  async_g2l_b128<32>(sK, mk);
  async_g2l_b128<48>(sK, mk);
  async_g2l_b128<0>(sV, mv);
  async_g2l_b128<16>(sV, mv);
  async_g2l_b128<32>(sV, mv);
  async_g2l_b128<48>(sV, mv);
  wait_async0();  // this wave's async LDS writes are complete
  unsigned short* dK = out_knew + (size_t)q * 32;
  unsigned short* dV = out_vnew + (size_t)q * 32;
  async_l2g_b128<0>(dK, mk);
  async_l2g_b128<16>(dK, mk);
  async_l2g_b128<32>(dK, mk);
  async_l2g_b128<48>(dK, mk);
  async_l2g_b128<0>(dV, mv);
  async_l2g_b128<16>(dV, mv);
  async_l2g_b128<32>(dV, mv);
  async_l2g_b128<48>(dV, mv);
  // S_ENDPGM performs an implicit wait-idle, draining ASYNCcnt stores.
}

// ---------------------------------------------------------------------------
// Kernel C: keys/values = gather prefix [:curr_pos+seq_len] with
// repeat_interleave of heads by n_rep. One thread per 64B output quad.
// ---------------------------------------------------------------------------
__global__ void __launch_bounds__(256)
kv_gather_rep(const unsigned short* __restrict__ kc,
              const unsigned short* __restrict__ vc,
              const unsigned short* __restrict__ xk,
              const unsigned short* __restrict__ xv,
              const int* __restrict__ Lp,
              const int* __restrict__ Pp,
              const int* __restrict__ Rp,
              int seq_len,
              unsigned short* __restrict__ out_keys,
              unsigned short* __restrict__ out_vals,
              int nQuads, int quadsPerB) {
  const int q = blockIdx.x * 256 + threadIdx.x;
  if (q >= nQuads) return;

  const int L = *Lp;
  const int P = *Pp;
  const int R = *Rp;

  const int b   = q / quadsPerB;
  const int rem = q - b * quadsPerB;
  const int dq  = rem & 3;             // 64B quad within head row
  const int row = rem >> 2;            // row = t * (kNKV*R) + hh
  const int H   = kNKV * R;
  const int t   = row / H;
  const int hh  = row - t * H;
  const int h   = hh / R;              // kv head after repeat_interleave

  const unsigned short* sK;
  const unsigned short* sV;
  if (t >= P && t < P + seq_len) {
    const size_t xo =
        ((((size_t)b * seq_len + (size_t)(t - P)) * kNKV + h) * kHD) +
        (size_t)dq * 32;
    sK = xk + xo;
    sV = xv + xo;
  } else {
    const size_t co =
        (((((size_t)L * kBSZ + b) * kMaxSeq + t) * kNKV + h) * kHD) +
        (size_t)dq * 32;
    sK = kc + co;
    sV = vc + co;
  }

  const uint4* sK4 = reinterpret_cast<const uint4*>(sK);
  const uint4* sV4 = reinterpret_cast<const uint4*>(sV);
  uint4* dK4 = reinterpret_cast<uint4*>(out_keys + (size_t)q * 32);
  uint4* dV4 = reinterpret_cast<uint4*>(out_vals + (size_t)q * 32);
#pragma unroll
  for (int i = 0; i < 4; ++i) dK4[i] = sK4[i];
#pragma unroll
  for (int i = 0; i < 4; ++i) dV4[i] = sV4[i];
}

// ---------------------------------------------------------------------------
// Host launcher.
// Input order: k_cache, v_cache, xk, xv, layer_idx, curr_pos, n_rep
// Output order: keys, values, k_new, v_new (flat bf16, concatenated)
// ---------------------------------------------------------------------------
extern "C" void kernel_launch(void* const* d_in, const int* in_sizes, int n_in,
                              void* d_out, int out_size, void* d_ws, size_t ws_size,
                              hipStream_t stream) {
  (void)n_in; (void)d_ws; (void)ws_size;

  const unsigned short* kc = (const unsigned short*)d_in[0];
  const unsigned short* vc = (const unsigned short*)d_in[1];
  const unsigned short* xk = (const unsigned short*)d_in[2];
  const unsigned short* xv = (const unsigned short*)d_in[3];
  const int* Lp = (const int*)d_in[4];
  const int* Pp = (const int*)d_in[5];
  const int* Rp = (const int*)d_in[6];

  const long long cacheElems = (long long)in_sizes[0];                 // 134,217,728
  const int seq_len = in_sizes[2] / (kBSZ * kNKV * kHD);               // 16
  const long long keysElems = ((long long)out_size - 2 * cacheElems) / 2;

  unsigned short* out_keys = (unsigned short*)d_out;
  unsigned short* out_vals = out_keys + keysElems;
  unsigned short* out_knew = out_vals + keysElems;
  unsigned short* out_vnew = out_knew + cacheElems;

  const int nQuadsA = (int)(cacheElems / 32);          // 4,194,304 threads
  kv_cache_writeback<<<dim3((unsigned)((nQuadsA + 255) / 256)), dim3(256), 0, stream>>>(
      kc, vc, xk, xv, Lp, Pp, seq_len, out_knew, out_vnew, nQuadsA);

  const int nQuadsC  = (int)(keysElems / 32);          // 528,384 threads
  const int quadsPerB = (int)(keysElems / (32 * kBSZ));
  kv_gather_rep<<<dim3((unsigned)((nQuadsC + 255) / 256)), dim3(256), 0, stream>>>(
      kc, vc, xk, xv, Lp, Pp, Rp, seq_len, out_keys, out_vals, nQuadsC, quadsPerB);
}